// View_selector_Critical_34961033789530
// MI455X (gfx1250) — compile-verified
//
#include <hip/hip_runtime.h>
#include <hip/hip_bf16.h>

#define NB_N   20          // views
#define NB_C   4096        // channels
#define NB_S   7           // sampled views
#define BLOCK  256
#define VPT    4           // float4 groups per thread (4 * 4 * 256 = 4096 channels)

#define ASYNC_OK (__has_builtin(__builtin_amdgcn_global_load_async_to_lds_b128) && \
                  __has_builtin(__builtin_amdgcn_global_store_async_from_lds_b128) && \
                  __has_builtin(__builtin_amdgcn_s_wait_asynccnt))

#if ASYNC_OK
typedef int v4i __attribute__((vector_size(16)));
typedef __attribute__((address_space(1))) v4i glb_v4i;   // global side of async DMA
typedef __attribute__((address_space(3))) v4i lds_v4i;   // LDS side of async DMA
#endif

__global__ __launch_bounds__(BLOCK) void
view_select_gather(const float* __restrict__ F0, const float* __restrict__ V0,
                   float* __restrict__ outF, float* __restrict__ outV) {
  const int b   = blockIdx.x;
  const int tid = threadIdx.x;

  __shared__ int   counts[NB_N];
  __shared__ int   idx_s[NB_S];
  __shared__ float stage[2][NB_C];   // 2 x 16KB double buffer for async DMA gather

  if (tid < NB_N) counts[tid] = 0;
  __syncthreads();

  const float*  Fb  = F0 + (size_t)b * NB_N * NB_C;
  const float4* Fb4 = (const float4*)Fb;            // [NB_N][NB_C/4]

  // ---- per-channel argmax over the view axis, float4-wide (b128 loads) ----
  // thread owns channels {k*1024 + tid*4 .. +3} for k = 0..3 (fully coalesced)
  float4 best_v[VPT];
  int4   best_n[VPT];
#pragma unroll
  for (int k = 0; k < VPT; ++k) {
    best_v[k] = Fb4[k * BLOCK + tid];               // n = 0
    best_n[k] = make_int4(0, 0, 0, 0);
  }
  for (int n = 1; n < NB_N; ++n) {
    const float4* row = Fb4 + n * (NB_C / 4);
#pragma unroll
    for (int k = 0; k < VPT; ++k) {
      const float4 v = row[k * BLOCK + tid];
      if (v.x > best_v[k].x) { best_v[k].x = v.x; best_n[k].x = n; }
      if (v.y > best_v[k].y) { best_v[k].y = v.y; best_n[k].y = n; }
      if (v.z > best_v[k].z) { best_v[k].z = v.z; best_n[k].z = n; }
      if (v.w > best_v[k].w) { best_v[k].w = v.w; best_n[k].w = n; }
    }
  }
  // ---- histogram into 20 LDS bins (ds_add atomics) ----
#pragma unroll
  for (int k = 0; k < VPT; ++k) {
    atomicAdd(&counts[best_n[k].x], 1);
    atomicAdd(&counts[best_n[k].y], 1);
    atomicAdd(&counts[best_n[k].z], 1);
    atomicAdd(&counts[best_n[k].w], 1);
  }
  __syncthreads();

  // ---- view selection (trivial 20-bin scan, one thread; keep code compact) ----
  if (tid == 0) {
    int u = 0;
#pragma clang loop unroll(disable)
    for (int n = 0; n < NB_N; ++n) u += (counts[n] > 0) ? 1 : 0;
    if (u >= NB_S) {
      // top-7 by count, ties -> lower view id (lax.top_k semantics)
      unsigned used = 0u;
#pragma clang loop unroll(disable)
      for (int j = 0; j < NB_S; ++j) {
        int bi = 0, bc = -1;
#pragma clang loop unroll(disable)
        for (int n = 0; n < NB_N; ++n)
          if (!((used >> n) & 1u) && counts[n] > bc) { bc = counts[n]; bi = n; }
        used |= 1u << bi;
        idx_s[j] = bi;
      }
    } else {
      // ascending nonzero bins, left-padded with the smallest
      int uniq[NB_S];
      int m = 0;
#pragma clang loop unroll(disable)
      for (int n = 0; n < NB_N; ++n)
        if (counts[n] > 0 && m < NB_S) uniq[m++] = n;
#pragma clang loop unroll(disable)
      for (int j = 0; j < NB_S; ++j) {
        int p = j - (NB_S - u);
        if (p < 0) p = 0;
        idx_s[j] = uniq[p];
      }
    }
  }
  __syncthreads();

  // ---- vertices gather: 7 * 3 floats ----
  if (tid < NB_S * 3) {
    const int s = tid / 3, d = tid % 3;
    outV[(size_t)b * NB_S * 3 + tid] = V0[(size_t)b * NB_N * 3 + idx_s[s] * 3 + d];
  }

  // ---- F gather: 7 rows x 16KB each ----
#if ASYNC_OK
  // CDNA5 async DMA path: global -> LDS -> global, never touching VGPR data.
  // Each wave owns a private 2KB slice of the 16KB row buffer; double-buffered
  // across rows so stores of row s drain while loads of row s+1 fill.
  const int w     = tid >> 5;           // wave id, 0..7
  const int lane  = tid & 31;
  const int wbase = w * (NB_C / 8);     // 512 floats = 2KB per wave

  for (int s = 0; s < NB_S; ++s) {
    const float* src = Fb + idx_s[s] * NB_C;
    float*       dst = outF + ((size_t)b * NB_S + s) * NB_C;
    float*       buf = stage[s & 1];
#pragma unroll
    for (int i = 0; i < 4; ++i) {
      const int off = wbase + i * 128 + lane * 4;   // B128 per lane
      __builtin_amdgcn_global_load_async_to_lds_b128(
          (glb_v4i*)(src + off), (lds_v4i*)(buf + off), 0, 0);
    }
    // loads done (and, transitively, last round's stores) before reading LDS
    __builtin_amdgcn_s_wait_asynccnt(0);
#pragma unroll
    for (int i = 0; i < 4; ++i) {
      const int off = wbase + i * 128 + lane * 4;
      __builtin_amdgcn_global_store_async_from_lds_b128(
          (glb_v4i*)(dst + off), (lds_v4i*)(buf + off), 0, 0);
    }
  }
  __builtin_amdgcn_s_wait_asynccnt(0);
#else
  // Fallback: plain vectorized copy (rows are L2-resident)
  for (int s = 0; s < NB_S; ++s) {
    const float4* src = (const float4*)(Fb + idx_s[s] * NB_C);
    float4*       dst = (float4*)(outF + ((size_t)b * NB_S + s) * NB_C);
    for (int i = tid; i < NB_C / 4; i += BLOCK) dst[i] = src[i];
  }
  (void)stage;
#endif
}

extern "C" void kernel_launch(void* const* d_in, const int* in_sizes, int n_in,
                              void* d_out, int out_size, void* d_ws, size_t ws_size,
                              hipStream_t stream) {
  const float* F0 = (const float*)d_in[0];
  const float* V0 = (const float*)d_in[1];
  const int B = in_sizes[0] / (NB_N * NB_C);   // 256

  float* outF = (float*)d_out;                        // [B, 7, 4096]
  float* outV = outF + (size_t)B * NB_S * NB_C;       // [B, 7, 3]

  view_select_gather<<<B, BLOCK, 0, stream>>>(F0, V0, outF, outV);
}